// NTM_58669253263745
// MI455X (gfx1250) — compile-verified
//
#include <hip/hip_runtime.h>
#include <hip/hip_bf16.h>
#include <math.h>

// ---------------------------------------------------------------------------
// NTM cell, fused, MI455X (gfx1250).  wave32.  WMMA f32 16x16x4 for the MLP.
// ---------------------------------------------------------------------------

typedef __attribute__((ext_vector_type(2))) float v2f;
typedef __attribute__((ext_vector_type(8))) float v8f;

#define NWAVES 4          // waves per block
#define ITEMS_PER_WAVE 16 // one WMMA M-tile
#define ITEMS_PER_BLOCK (NWAVES * ITEMS_PER_WAVE)

__device__ __forceinline__ float sigm(float x) { return 1.0f / (1.0f + __expf(-x)); }
__device__ __forceinline__ float softplus_f(float x) {
  return fmaxf(x, 0.0f) + log1pf(__expf(-fabsf(x)));
}
__device__ __forceinline__ float tanh_fast(float x) {
  float e = __expf(-2.0f * fabsf(x));
  float t = (1.0f - e) / (1.0f + e);
  return copysignf(t, x);
}

__device__ __forceinline__ v8f wmma4(v2f a, v2f b, v8f c) {
  // D(16x16,f32) += A(16x4,f32) * B(4x16,f32)   -- v_wmma_f32_16x16x4_f32
  return __builtin_amdgcn_wmma_f32_16x16x4_f32(false, a, false, b, (short)0, c,
                                               false, false);
}

__global__ __launch_bounds__(128) void ntm_fused(
    const float* __restrict__ inputs,      // B x 8
    const float* __restrict__ prev_reads,  // B x 8
    const float* __restrict__ pwr,         // B x 100
    const float* __restrict__ pww,         // B x 100
    const float* __restrict__ mem,         // B x 100 x 8
    const float* __restrict__ W1,          // 16 x 128
    const float* __restrict__ b1,          // 128
    const float* __restrict__ Wi,          // 128 x 32
    const float* __restrict__ bi,          // 32
    const float* __restrict__ We,          // 128 x 8
    const float* __restrict__ be,          // 8
    const float* __restrict__ Wr,          // 32 x 14
    const float* __restrict__ br,          // 14
    const float* __restrict__ Ww,          // 32 x 30
    const float* __restrict__ bw,          // 30
    float* __restrict__ out,               // B x 816
    long Btot) {
  // ---- shared memory -------------------------------------------------------
  __shared__ __align__(16) float sW1[16 * 128];
  __shared__ __align__(16) float sb1[128];
  __shared__ __align__(16) float sWi[128 * 32];
  __shared__ __align__(16) float sbi[32];
  __shared__ __align__(16) float sWe[128 * 16];  // padded N 8->16
  __shared__ __align__(16) float sbe[16];
  __shared__ __align__(16) float sWr[32 * 16];   // padded N 14->16
  __shared__ __align__(16) float sbr[16];
  __shared__ __align__(16) float sWw[32 * 32];   // padded N 30->32
  __shared__ __align__(16) float sbw[32];
  __shared__ __align__(16) float sX[NWAVES][16 * 16];
  __shared__ __align__(16) float sH[NWAVES][16 * 128];
  __shared__ __align__(16) float sI[NWAVES][16 * 32];
  __shared__ __align__(16) float sPR[NWAVES][16 * 16];
  __shared__ __align__(16) float sPW[NWAVES][16 * 32];
  __shared__ __align__(16) float sEXT[NWAVES][16 * 8];

  const int tid = threadIdx.x;
  const int w = tid >> 5;
  const int lane = tid & 31;
  const long bblk = (long)blockIdx.x * ITEMS_PER_BLOCK;
  if (bblk >= Btot) return;  // block-uniform

  // ---- stage weights into LDS (block cooperative) --------------------------
  for (int i = tid; i < 16 * 128; i += 128) sW1[i] = W1[i];
  if (tid < 128) sb1[tid] = b1[tid];
  for (int i = tid; i < 128 * 32; i += 128) sWi[i] = Wi[i];
  if (tid < 32) sbi[tid] = bi[tid];
  for (int i = tid; i < 128 * 16; i += 128) {
    int k = i >> 4, n = i & 15;
    sWe[i] = (n < 8) ? We[k * 8 + n] : 0.0f;
  }
  if (tid < 16) sbe[tid] = (tid < 8) ? be[tid] : 0.0f;
  for (int i = tid; i < 32 * 16; i += 128) {
    int k = i >> 4, n = i & 15;
    sWr[i] = (n < 14) ? Wr[k * 14 + n] : 0.0f;
  }
  if (tid < 16) sbr[tid] = (tid < 14) ? br[tid] : 0.0f;
  for (int i = tid; i < 32 * 32; i += 128) {
    int k = i >> 5, n = i & 31;
    sWw[i] = (n < 30) ? Ww[k * 30 + n] : 0.0f;
  }
  if (tid < 32) sbw[tid] = (tid < 30) ? bw[tid] : 0.0f;

  // ---- stage X = [inputs | prev_reads] for this wave's 16 items ------------
  const long b0 = bblk + (long)w * ITEMS_PER_WAVE;
  {
    float* Xw = sX[w];
    if (lane < 16) {
      const float4* p = (const float4*)(inputs + (b0 + lane) * 8);
      float4 a = p[0], c = p[1];
      float4* q = (float4*)&Xw[lane * 16];
      q[0] = a;
      q[1] = c;
    } else {
      int m = lane - 16;
      const float4* p = (const float4*)(prev_reads + (b0 + m) * 8);
      float4 a = p[0], c = p[1];
      float4* q = (float4*)&Xw[m * 16 + 8];
      q[0] = a;
      q[1] = c;
    }
  }
  __syncthreads();

  float* Xw = sX[w];
  float* Hw = sH[w];
  float* Iw = sI[w];
  float* PRw = sPR[w];
  float* PWw = sPW[w];
  float* Ew = sEXT[w];

  const int col = lane & 15;  // N column (B/C/D) and M row (A) for this lane
  const int hi = lane >> 4;   // half-wave selector

  // ---- GEMM1: H = relu(X(16x16) @ W1(16x128) + b1) -------------------------
#pragma unroll
  for (int j = 0; j < 8; ++j) {
    v8f acc = {0, 0, 0, 0, 0, 0, 0, 0};
#pragma unroll
    for (int t = 0; t < 4; ++t) {
      int k0 = 4 * t + 2 * hi;
      v2f a, bb;
      a.x = Xw[col * 16 + k0];
      a.y = Xw[col * 16 + k0 + 1];
      bb.x = sW1[k0 * 128 + j * 16 + col];
      bb.y = sW1[(k0 + 1) * 128 + j * 16 + col];
      acc = wmma4(a, bb, acc);
    }
    float bias = sb1[j * 16 + col];
#pragma unroll
    for (int v = 0; v < 8; ++v)
      Hw[(v + 8 * hi) * 128 + j * 16 + col] = fmaxf(acc[v] + bias, 0.0f);
  }

  // ---- GEMM2: internal = tanh(H(16x128) @ Wi(128x32) + bi) -----------------
#pragma unroll
  for (int j = 0; j < 2; ++j) {
    v8f acc = {0, 0, 0, 0, 0, 0, 0, 0};
#pragma unroll
    for (int t = 0; t < 32; ++t) {
      int k0 = 4 * t + 2 * hi;
      v2f a, bb;
      a.x = Hw[col * 128 + k0];
      a.y = Hw[col * 128 + k0 + 1];
      bb.x = sWi[k0 * 32 + j * 16 + col];
      bb.y = sWi[(k0 + 1) * 32 + j * 16 + col];
      acc = wmma4(a, bb, acc);
    }
    float bias = sbi[j * 16 + col];
#pragma unroll
    for (int v = 0; v < 8; ++v)
      Iw[(v + 8 * hi) * 32 + j * 16 + col] = tanh_fast(acc[v] + bias);
  }

  // ---- GEMM3: external = sigmoid(H @ We(128x8,padded16) + be) --------------
  {
    v8f acc = {0, 0, 0, 0, 0, 0, 0, 0};
#pragma unroll
    for (int t = 0; t < 32; ++t) {
      int k0 = 4 * t + 2 * hi;
      v2f a, bb;
      a.x = Hw[col * 128 + k0];
      a.y = Hw[col * 128 + k0 + 1];
      bb.x = sWe[k0 * 16 + col];
      bb.y = sWe[(k0 + 1) * 16 + col];
      acc = wmma4(a, bb, acc);
    }
    float bias = sbe[col];
#pragma unroll
    for (int v = 0; v < 8; ++v)
      if (col < 8) Ew[(v + 8 * hi) * 8 + col] = sigm(acc[v] + bias);
  }

  // ---- GEMM4: pr = internal(16x32) @ Wr(32x14,padded16) + br ---------------
  {
    v8f acc = {0, 0, 0, 0, 0, 0, 0, 0};
#pragma unroll
    for (int t = 0; t < 8; ++t) {
      int k0 = 4 * t + 2 * hi;
      v2f a, bb;
      a.x = Iw[col * 32 + k0];
      a.y = Iw[col * 32 + k0 + 1];
      bb.x = sWr[k0 * 16 + col];
      bb.y = sWr[(k0 + 1) * 16 + col];
      acc = wmma4(a, bb, acc);
    }
    float bias = sbr[col];
#pragma unroll
    for (int v = 0; v < 8; ++v)
      PRw[(v + 8 * hi) * 16 + col] = acc[v] + bias;
  }

  // ---- GEMM5: pw = internal @ Ww(32x30,padded32) + bw ----------------------
#pragma unroll
  for (int j = 0; j < 2; ++j) {
    v8f acc = {0, 0, 0, 0, 0, 0, 0, 0};
#pragma unroll
    for (int t = 0; t < 8; ++t) {
      int k0 = 4 * t + 2 * hi;
      v2f a, bb;
      a.x = Iw[col * 32 + k0];
      a.y = Iw[col * 32 + k0 + 1];
      bb.x = sWw[k0 * 32 + j * 16 + col];
      bb.y = sWw[(k0 + 1) * 32 + j * 16 + col];
      acc = wmma4(a, bb, acc);
    }
    float bias = sbw[j * 16 + col];
#pragma unroll
    for (int v = 0; v < 8; ++v)
      PWw[(v + 8 * hi) * 32 + j * 16 + col] = acc[v] + bias;
  }

  // ==== addressing phase: lane pair per item, 50 slots per lane =============
  const int item = lane >> 1;
  const int h = lane & 1;
  const long b = b0 + item;
  const float* prrow = &PRw[item * 16];
  const float* pwrow = &PWw[item * 32];

  float kR[8], kW[8], er[8], ad[8];
#pragma unroll
  for (int m = 0; m < 8; ++m) {
    kR[m] = tanh_fast(prrow[m]);
    kW[m] = tanh_fast(pwrow[m]);
    er[m] = sigm(pwrow[14 + m]);
    ad[m] = pwrow[22 + m];
  }
  float betaR = softplus_f(prrow[8]);
  float gR = sigm(prrow[9]);
  float gammaR = 1.0f + softplus_f(prrow[13]);
  float betaW = softplus_f(pwrow[8]);
  float gW = sigm(pwrow[9]);
  float gammaW = 1.0f + softplus_f(pwrow[13]);
  // shift softmaxes (3-way, stable)
  float s0R, s1R, s2R, s0W, s1W, s2W;
  {
    float a0 = prrow[10], a1 = prrow[11], a2 = prrow[12];
    float m3 = fmaxf(a0, fmaxf(a1, a2));
    float e0 = __expf(a0 - m3), e1 = __expf(a1 - m3), e2 = __expf(a2 - m3);
    float inv = 1.0f / (e0 + e1 + e2);
    s0R = e0 * inv; s1R = e1 * inv; s2R = e2 * inv;
  }
  {
    float a0 = pwrow[10], a1 = pwrow[11], a2 = pwrow[12];
    float m3 = fmaxf(a0, fmaxf(a1, a2));
    float e0 = __expf(a0 - m3), e1 = __expf(a1 - m3), e2 = __expf(a2 - m3);
    float inv = 1.0f / (e0 + e1 + e2);
    s0W = e0 * inv; s1W = e1 * inv; s2W = e2 * inv;
  }
  float knR = 0.0f, knW = 0.0f;
#pragma unroll
  for (int m = 0; m < 8; ++m) {
    knR += kR[m] * kR[m];
    knW += kW[m] * kW[m];
  }
  knR = sqrtf(knR);
  knW = sqrtf(knW);

  const float* memb = mem + (size_t)b * 800;
  const float* prow_r = pwr + (size_t)b * 100 + 50 * h;
  const float* prow_w = pww + (size_t)b * 100 + 50 * h;
  __builtin_prefetch(prow_r, 0, 1);
  __builtin_prefetch(prow_w, 0, 1);

  float simR[50], simW[50];

  // ---- pass 1 over memory: cosine similarities (both heads) ----------------
#pragma unroll
  for (int j = 0; j < 50; ++j) {
    int n = 50 * h + j;
    float4 m0 = *(const float4*)(memb + n * 8);
    float4 m1 = *(const float4*)(memb + n * 8 + 4);
    float dR = kR[0] * m0.x + kR[1] * m0.y + kR[2] * m0.z + kR[3] * m0.w +
               kR[4] * m1.x + kR[5] * m1.y + kR[6] * m1.z + kR[7] * m1.w;
    float dW = kW[0] * m0.x + kW[1] * m0.y + kW[2] * m0.z + kW[3] * m0.w +
               kW[4] * m1.x + kW[5] * m1.y + kW[6] * m1.z + kW[7] * m1.w;
    float nsq = m0.x * m0.x + m0.y * m0.y + m0.z * m0.z + m0.w * m0.w +
                m1.x * m1.x + m1.y * m1.y + m1.z * m1.z + m1.w * m1.w;
    float nm = sqrtf(nsq);
    simR[j] = dR / (nm * knR + 1e-8f);
    simW[j] = dW / (nm * knW + 1e-8f);
  }

  // ---- sim -> final addressing weight (softmax, interpolate, shift, pow) ---
  auto addr_weights = [&](float(&s)[50], float beta, float g, float c0, float c1,
                          float c2, float gamma, const float* prevw) {
    float mx = -1e30f;
#pragma unroll
    for (int j = 0; j < 50; ++j) mx = fmaxf(mx, s[j]);
    mx = fmaxf(mx, __shfl_xor(mx, 1, 32));
    float sum = 0.0f;
#pragma unroll
    for (int j = 0; j < 50; ++j) {
      float e = __expf(beta * (s[j] - mx));
      s[j] = e;
      sum += e;
    }
    sum += __shfl_xor(sum, 1, 32);
    float inv = 1.0f / sum;
#pragma unroll
    for (int j = 0; j < 50; ++j) {
      float wc = s[j] * inv;
      s[j] = g * wc + (1.0f - g) * prevw[j];
    }
    // circular shift conv: w_s[n] = c0*w[n+1] + c1*w[n] + c2*w[n-1]
    float nb0 = __shfl_xor(s[0], 1, 32);    // partner's first slot
    float nb49 = __shfl_xor(s[49], 1, 32);  // partner's last slot
    float prevOld = nb49;
    float psum = 0.0f;
#pragma unroll
    for (int j = 0; j < 50; ++j) {
      float cur = s[j];
      float nxt = (j < 49) ? s[(j + 1) % 50] : nb0;
      float ws = c0 * nxt + c1 * cur + c2 * prevOld;
      float wp = exp2f(gamma * __log2f(ws));  // ws^gamma, ws >= 0
      s[j] = wp;
      psum += wp;
      prevOld = cur;
    }
    psum += __shfl_xor(psum, 1, 32);
    float invp = 1.0f / (psum + 1e-8f);
#pragma unroll
    for (int j = 0; j < 50; ++j) s[j] *= invp;
  };

  addr_weights(simR, betaR, gR, s0R, s1R, s2R, gammaR, prow_r);
  addr_weights(simW, betaW, gW, s0W, s1W, s2W, gammaW, prow_w);

  // ---- pass 2 over memory: read vector + memory update + stores ------------
  float* outb = out + (size_t)b * 816;
  float rd[8] = {0, 0, 0, 0, 0, 0, 0, 0};
#pragma unroll
  for (int j = 0; j < 50; ++j) {
    int n = 50 * h + j;
    float4 m0 = *(const float4*)(memb + n * 8);
    float4 m1 = *(const float4*)(memb + n * 8 + 4);
    float wr = simR[j];
    float wt = simW[j];
    rd[0] += wr * m0.x; rd[1] += wr * m0.y; rd[2] += wr * m0.z; rd[3] += wr * m0.w;
    rd[4] += wr * m1.x; rd[5] += wr * m1.y; rd[6] += wr * m1.z; rd[7] += wr * m1.w;
    float4 o0, o1;
    o0.x = m0.x * (1.0f - wt * er[0]) + wt * ad[0];
    o0.y = m0.y * (1.0f - wt * er[1]) + wt * ad[1];
    o0.z = m0.z * (1.0f - wt * er[2]) + wt * ad[2];
    o0.w = m0.w * (1.0f - wt * er[3]) + wt * ad[3];
    o1.x = m1.x * (1.0f - wt * er[4]) + wt * ad[4];
    o1.y = m1.y * (1.0f - wt * er[5]) + wt * ad[5];
    o1.z = m1.z * (1.0f - wt * er[6]) + wt * ad[6];
    o1.w = m1.w * (1.0f - wt * er[7]) + wt * ad[7];
    float* op = outb + 16 + n * 8;
    *(float4*)op = o0;
    *(float4*)(op + 4) = o1;
  }
#pragma unroll
  for (int m = 0; m < 8; ++m) rd[m] += __shfl_xor(rd[m], 1, 32);
  float4 rv;
  rv.x = h ? rd[4] : rd[0];
  rv.y = h ? rd[5] : rd[1];
  rv.z = h ? rd[6] : rd[2];
  rv.w = h ? rd[7] : rd[3];
  *(float4*)(outb + 8 + 4 * h) = rv;
  float4 ev = *(const float4*)&Ew[item * 8 + 4 * h];
  *(float4*)(outb + 4 * h) = ev;
}

extern "C" void kernel_launch(void* const* d_in, const int* in_sizes, int n_in,
                              void* d_out, int out_size, void* d_ws,
                              size_t ws_size, hipStream_t stream) {
  (void)n_in; (void)out_size; (void)d_ws; (void)ws_size;
  const float* inputs = (const float*)d_in[0];
  const float* prev_reads = (const float*)d_in[1];
  const float* pwr = (const float*)d_in[2];
  const float* pww = (const float*)d_in[3];
  const float* mem = (const float*)d_in[4];
  const float* W1 = (const float*)d_in[5];
  const float* b1 = (const float*)d_in[6];
  const float* Wi = (const float*)d_in[7];
  const float* bi = (const float*)d_in[8];
  const float* We = (const float*)d_in[9];
  const float* be = (const float*)d_in[10];
  const float* Wr = (const float*)d_in[11];
  const float* br = (const float*)d_in[12];
  const float* Ww = (const float*)d_in[13];
  const float* bw = (const float*)d_in[14];
  float* outp = (float*)d_out;

  long B = (long)in_sizes[0] / 8;
  int nblocks = (int)((B + ITEMS_PER_BLOCK - 1) / ITEMS_PER_BLOCK);
  ntm_fused<<<nblocks, 128, 0, stream>>>(inputs, prev_reads, pwr, pww, mem, W1,
                                         b1, Wi, bi, We, be, Wr, br, Ww, bw,
                                         outp, B);
}